// gnn_torch_module_39453569581395
// MI455X (gfx1250) — compile-verified
//
#include <hip/hip_runtime.h>
#include <hip/hip_bf16.h>
#include <math.h>

#define NN 50000
#define NE 800000
#define ENC 64
#define ROUNDS 3
#define OUTD 5
#define NEG_SLOPE 0.2f

typedef float v2f __attribute__((ext_vector_type(2)));
typedef float v8f __attribute__((ext_vector_type(8)));

// ---------------------------------------------------------------- utilities
__device__ __forceinline__ void atomicMaxF(float* addr, float v) {
  // float max via monotone integer atomics (valid incl. -inf init)
  if (v >= 0.0f) atomicMax((int*)addr, __float_as_int(v));
  else           atomicMin((unsigned int*)addr, __float_as_uint(v));
}

// ---------------------------------------------------- node encoder: h0 = x@Wn + bn
__global__ __launch_bounds__(256) void k_node_enc(
    const float* __restrict__ x, const float* __restrict__ Wn,
    const float* __restrict__ bn, float* __restrict__ h0) {
  int i = blockIdx.x * 256 + threadIdx.x;
  if (i >= NN * ENC) return;
  int n = i >> 6, j = i & 63;
  const float* xr = x + n * 8;
  float acc = bn[j];
#pragma unroll
  for (int k = 0; k < 8; ++k) acc += xr[k] * Wn[k * ENC + j];
  h0[i] = acc;
}

// ---------------------------------------- fold edge encoder into round weights
// Wc[4,64] = W_edge @ We ; bc[64] = b_edge @ We   (one block per (stack,round))
__global__ __launch_bounds__(256) void k_fold(
    const float* __restrict__ We, const float* __restrict__ W_edge,
    const float* __restrict__ b_edge, float* __restrict__ Wc,
    float* __restrict__ bc) {
  __shared__ float lWe[ENC * ENC];
  int tid = threadIdx.x;
  for (int i = tid; i < ENC * ENC; i += 256) lWe[i] = We[i];
  __syncthreads();
  int k = tid >> 6, j = tid & 63;
  float acc = 0.f;
  for (int q = 0; q < ENC; ++q) acc += W_edge[k * ENC + q] * lWe[q * ENC + j];
  Wc[tid] = acc;
  if (k == 0) {
    float bb = 0.f;
    for (int q = 0; q < ENC; ++q) bb += b_edge[q] * lWe[q * ENC + j];
    bc[j] = bb;
  }
}

// --------------------------------------------- WMMA GEMM: C[M,64] = A[M,64] @ W[64,64]
// 256 threads = 8 waves; each wave owns one 16-row strip. K=64 via 16 steps of
// V_WMMA_F32_16X16X4_F32, 4 N-tiles of 16.
__global__ __launch_bounds__(256) void k_gemm64(
    const float* __restrict__ A, const float* __restrict__ W,
    float* __restrict__ C, int M) {
  __shared__ float lw[ENC * ENC];       // 16 KB weights
  __shared__ float la[8][16 * ENC];     // 8 waves * 4 KB A strips
  int tid = threadIdx.x;
  for (int i = tid; i < ENC * ENC; i += 256) lw[i] = W[i];
  __syncthreads();

  int wave = tid >> 5, lane = tid & 31;
  int strip = blockIdx.x * 8 + wave;
  int nstrips = (M + 15) >> 4;          // M is a multiple of 16 here
  if (strip >= nstrips) return;
  int row0 = strip << 4;

  // stage A strip (contiguous 1024 floats) — same-wave LDS ops are in-order
  const float* Ab = A + (size_t)row0 * ENC;
  float* las = la[wave];
  for (int i = lane; i < 16 * ENC; i += 32) las[i] = Ab[i];

  int half = lane >> 4;   // 0: K pair {0,1}, 1: K pair {2,3}
  int l16  = lane & 15;
  v8f acc0 = {}, acc1 = {}, acc2 = {}, acc3 = {};

  for (int k0 = 0; k0 < ENC; k0 += 4) {
    int kp = k0 + half * 2;
    v2f a;
    a.x = las[l16 * ENC + kp];
    a.y = las[l16 * ENC + kp + 1];
    v2f b0, b1, b2, b3;
    b0.x = lw[kp * ENC + 0  + l16]; b0.y = lw[(kp + 1) * ENC + 0  + l16];
    b1.x = lw[kp * ENC + 16 + l16]; b1.y = lw[(kp + 1) * ENC + 16 + l16];
    b2.x = lw[kp * ENC + 32 + l16]; b2.y = lw[(kp + 1) * ENC + 32 + l16];
    b3.x = lw[kp * ENC + 48 + l16]; b3.y = lw[(kp + 1) * ENC + 48 + l16];
    acc0 = __builtin_amdgcn_wmma_f32_16x16x4_f32(false, a, false, b0, (short)0, acc0, false, false);
    acc1 = __builtin_amdgcn_wmma_f32_16x16x4_f32(false, a, false, b1, (short)0, acc1, false, false);
    acc2 = __builtin_amdgcn_wmma_f32_16x16x4_f32(false, a, false, b2, (short)0, acc2, false, false);
    acc3 = __builtin_amdgcn_wmma_f32_16x16x4_f32(false, a, false, b3, (short)0, acc3, false, false);
  }

  float* Cb = C + (size_t)row0 * ENC;
#pragma unroll
  for (int v = 0; v < 8; ++v) {
    int m = v + half * 8;
    Cb[m * ENC + 0  + l16] = acc0[v];
    Cb[m * ENC + 16 + l16] = acc1[v];
    Cb[m * ENC + 32 + l16] = acc2[v];
    Cb[m * ENC + 48 + l16] = acc3[v];
  }
}

// ----------------------------------------------- per-round init (smax/denom/acc)
__global__ __launch_bounds__(256) void k_init(
    float* __restrict__ smax, float* __restrict__ denom,
    float* __restrict__ acc) {
  int i = blockIdx.x * 256 + threadIdx.x;
  if (i >= NN * ENC) return;
  acc[i] = 0.f;
  if (i < NN) { smax[i] = -3.402823466e38f; denom[i] = 0.f; }
}

// --------------------------- edge pass 1: scores s[e] + segment max via atomics
__global__ __launch_bounds__(256) void k_edge_pass1(
    const float* __restrict__ ea, const int* __restrict__ src,
    const int* __restrict__ dst, const float* __restrict__ xl,
    const float* __restrict__ xr, const float* __restrict__ Wc,
    const float* __restrict__ bc, const float* __restrict__ att,
    float* __restrict__ sbuf, float* __restrict__ smax) {
  __shared__ float lwc[4 * ENC], lbc[ENC], latt[ENC];
  int tid = threadIdx.x;
  lwc[tid & 255] = Wc[tid & 255];
  if (tid < ENC) { lbc[tid] = bc[tid]; latt[tid] = att[tid]; }
  __syncthreads();

  int e = blockIdx.x * 256 + tid;
  if (e >= NE) return;
  int sa = src[e], da = dst[e];
  const float4 av = *(const float4*)(ea + (size_t)e * 4);
  const float4* xls = (const float4*)(xl + (size_t)sa * ENC);
  const float4* xrs = (const float4*)(xr + (size_t)da * ENC);
  float acc = 0.f;
#pragma unroll 4
  for (int q = 0; q < 16; ++q) {
    float4 vl = xls[q], vr = xrs[q];
    float mv[4] = {vl.x + vr.x, vl.y + vr.y, vl.z + vr.z, vl.w + vr.w};
#pragma unroll
    for (int c = 0; c < 4; ++c) {
      int j = 4 * q + c;
      float m = mv[c] + av.x * lwc[j] + av.y * lwc[ENC + j] +
                av.z * lwc[2 * ENC + j] + av.w * lwc[3 * ENC + j] + lbc[j];
      m = (m > 0.f) ? m : NEG_SLOPE * m;
      acc += latt[j] * m;
    }
  }
  sbuf[e] = acc;
  atomicMaxF(&smax[da], acc);
}

// ---- edge pass 2: ex = exp(s - smax[dst]); denom += ex; acc[dst] += ex*xl[src]
// 16 threads per edge -> float4-coalesced gather + f32 atomic scatter
__global__ __launch_bounds__(256) void k_edge_pass2(
    const int* __restrict__ src, const int* __restrict__ dst,
    const float* __restrict__ xl, const float* __restrict__ sbuf,
    const float* __restrict__ smax, float* __restrict__ denom,
    float* __restrict__ acc) {
  int tid = threadIdx.x;
  int t = tid & 15, el = tid >> 4;
  int e = blockIdx.x * 16 + el;
  if (e >= NE) return;
  int sa = src[e], da = dst[e];
  float ex = __expf(sbuf[e] - smax[da]);
  if (t == 0) atomicAdd(&denom[da], ex);
  float4 v = ((const float4*)(xl + (size_t)sa * ENC))[t];
  float* ob = acc + (size_t)da * ENC + t * 4;
  atomicAdd(ob + 0, ex * v.x);
  atomicAdd(ob + 1, ex * v.y);
  atomicAdd(ob + 2, ex * v.z);
  atomicAdd(ob + 3, ex * v.w);
}

// ------------------------------- finalize: h = acc/denom + b (+optional ReLU)
__global__ __launch_bounds__(256) void k_finalize(
    const float* __restrict__ acc, const float* __restrict__ denom,
    const float* __restrict__ b, float* __restrict__ hout, int applyRelu) {
  int i = blockIdx.x * 256 + threadIdx.x;
  if (i >= NN * ENC) return;
  int n = i >> 6, j = i & 63;
  float d = denom[n];
  float v = (d > 0.f) ? acc[i] / d : 0.f;
  v += b[j];
  if (applyRelu && v < 0.f) v = 0.f;
  hout[i] = v;
}

// ------------------------------------------------- decoders (actor / critic)
__global__ __launch_bounds__(256) void k_decode(
    const float* __restrict__ h, const float* __restrict__ W,
    const float* __restrict__ bias, float* __restrict__ out, int mode) {
  int n = blockIdx.x * 256 + threadIdx.x;
  if (n >= NN) return;
  const float* hr = h + (size_t)n * ENC;
  if (mode == 0) {
    float acc[OUTD];
#pragma unroll
    for (int a = 0; a < OUTD; ++a) acc[a] = bias[a];
    for (int k = 0; k < ENC; ++k) {
      float hv = hr[k];
#pragma unroll
      for (int a = 0; a < OUTD; ++a) acc[a] += hv * W[k * OUTD + a];
    }
#pragma unroll
    for (int a = 0; a < OUTD; ++a) out[(size_t)n * 6 + a] = tanhf(acc[a]);
  } else {
    float acc = bias[0];
    for (int k = 0; k < ENC; ++k) acc += hr[k] * W[k];
    out[(size_t)n * 6 + 5] = acc;
  }
}

// ---------------------------------------------------------------- launcher
extern "C" void kernel_launch(void* const* d_in, const int* in_sizes, int n_in,
                              void* d_out, int out_size, void* d_ws, size_t ws_size,
                              hipStream_t stream) {
  const float* x         = (const float*)d_in[0];
  const float* edge_attr = (const float*)d_in[1];
  const float* W_node    = (const float*)d_in[2];
  const float* b_node    = (const float*)d_in[3];
  const float* W_edge    = (const float*)d_in[4];
  const float* b_edge    = (const float*)d_in[5];
  const float* aWl  = (const float*)d_in[6];
  const float* aWr  = (const float*)d_in[7];
  const float* aWe  = (const float*)d_in[8];
  const float* aAtt = (const float*)d_in[9];
  const float* aB   = (const float*)d_in[10];
  const float* cWl  = (const float*)d_in[11];
  const float* cWr  = (const float*)d_in[12];
  const float* cWe  = (const float*)d_in[13];
  const float* cAtt = (const float*)d_in[14];
  const float* cB   = (const float*)d_in[15];
  const float* W_act = (const float*)d_in[16];
  const float* b_act = (const float*)d_in[17];
  const float* W_val = (const float*)d_in[18];
  const float* b_val = (const float*)d_in[19];
  const int*   src   = (const int*)d_in[20];
  const int*   dst   = (const int*)d_in[21];
  float* out = (float*)d_out;

  // workspace layout (floats)
  float* ws   = (float*)d_ws;
  size_t NH = (size_t)NN * ENC;
  float* h0    = ws;
  float* hcur  = h0 + NH;
  float* hnx   = hcur + NH;           // per-round accumulation buffer
  float* xl    = hnx + NH;
  float* xr    = xl + NH;
  float* sbuf  = xr + NH;             // NE
  float* smax  = sbuf + NE;           // NN
  float* denom = smax + NN;           // NN
  float* WcAll = denom + NN;          // 6 * 256
  float* bcAll = WcAll + 6 * 256;     // 6 * 64

  const int gNE64 = (NN * ENC) / 256;           // 12500
  const int gGemm = ((NN / 16) + 7) / 8;        // 391
  const int gE1   = NE / 256;                   // 3125
  const int gE2   = NE / 16;                    // 50000
  const int gDec  = (NN + 255) / 256;           // 196

  // encoders + folded edge weights
  k_node_enc<<<gNE64, 256, 0, stream>>>(x, W_node, b_node, h0);
  for (int s = 0; s < 2; ++s) {
    const float* We = s ? cWe : aWe;
    for (int r = 0; r < ROUNDS; ++r)
      k_fold<<<1, 256, 0, stream>>>(We + r * ENC * ENC, W_edge, b_edge,
                                    WcAll + (s * ROUNDS + r) * 256,
                                    bcAll + (s * ROUNDS + r) * ENC);
  }

  for (int s = 0; s < 2; ++s) {
    const float* Wl  = s ? cWl : aWl;
    const float* Wr  = s ? cWr : aWr;
    const float* att = s ? cAtt : aAtt;
    const float* bb  = s ? cB : aB;
    for (int r = 0; r < ROUNDS; ++r) {
      const float* hin = (r == 0) ? h0 : hcur;
      k_gemm64<<<gGemm, 256, 0, stream>>>(hin, Wl + r * ENC * ENC, xl, NN);
      k_gemm64<<<gGemm, 256, 0, stream>>>(hin, Wr + r * ENC * ENC, xr, NN);
      k_init<<<gNE64, 256, 0, stream>>>(smax, denom, hnx);
      k_edge_pass1<<<gE1, 256, 0, stream>>>(edge_attr, src, dst, xl, xr,
                                            WcAll + (s * ROUNDS + r) * 256,
                                            bcAll + (s * ROUNDS + r) * ENC,
                                            att + r * ENC, sbuf, smax);
      k_edge_pass2<<<gE2, 256, 0, stream>>>(src, dst, xl, sbuf, smax, denom, hnx);
      k_finalize<<<gNE64, 256, 0, stream>>>(hnx, denom, bb + r * ENC, hcur,
                                            (r < ROUNDS - 1) ? 1 : 0);
    }
    if (s == 0) k_decode<<<gDec, 256, 0, stream>>>(hcur, W_act, b_act, out, 0);
    else        k_decode<<<gDec, 256, 0, stream>>>(hcur, W_val, b_val, out, 1);
  }
}